// NeighborPoolingLayer_55490977465088
// MI455X (gfx1250) — compile-verified
//
#include <hip/hip_runtime.h>

// NeighborPoolingLayer: CSR segment-mean over gathered neighbor features.
//   out[m, :] = mean_{k in [splits[m], splits[m+1])} in_features[nbr_idx[k], :]
// N=1e6, C=64, M=131072, K=32 (uniform). Memory-bound gather-reduce
// (~1.07 GB random 256B-row gathers vs 268 MFLOP -> AI ~0.25 FLOP/B;
//  HBM floor ~50us @ 23.3 TB/s; 256MB table vs 192MB L2 -> partial reuse).
// Reduction done exactly in f32 via V_WMMA_F32_16X16X4_F32:
//   A[16x4] = gathered features (row = feature-in-16-block, col = neighbor),
//   B[4x16] = 1/count broadcast, D rows = features (replicated over columns).
// Indices/splits ride the scalar path (readfirstlane-uniformized). Gather
// loads are software double-buffered so each chunk's 8 loads overlap the
// previous chunk's WMMAs instead of stalling on s_wait_loadcnt 0.

typedef __attribute__((ext_vector_type(2))) float v2f;
typedef __attribute__((ext_vector_type(4))) float v4f;
typedef __attribute__((ext_vector_type(8))) float v8f;

#define WAVES_PER_BLOCK 8

#define WMMA4(x0, x1, x2, x3)                                                  \
    do {                                                                       \
        acc0 = __builtin_amdgcn_wmma_f32_16x16x4_f32(false, x0, false, bmat,   \
                                                     (short)0, acc0, false,    \
                                                     false);                   \
        acc1 = __builtin_amdgcn_wmma_f32_16x16x4_f32(false, x1, false, bmat,   \
                                                     (short)0, acc1, false,    \
                                                     false);                   \
        acc2 = __builtin_amdgcn_wmma_f32_16x16x4_f32(false, x2, false, bmat,   \
                                                     (short)0, acc2, false,    \
                                                     false);                   \
        acc3 = __builtin_amdgcn_wmma_f32_16x16x4_f32(false, x3, false, bmat,   \
                                                     (short)0, acc3, false,    \
                                                     false);                   \
    } while (0)

__global__ __launch_bounds__(256) void neighbor_pool_wmma(
    const float* __restrict__ feats,        // [N, 64] f32  (256 MB)
    const long long* __restrict__ nbr_idx,  // [E] i64
    const long long* __restrict__ splits,   // [M+1] i64
    float* __restrict__ out,                // [M, 64] f32
    int M)
{
    const int lane = threadIdx.x & 31;
    // readfirstlane: provably wave-uniform -> SGPRs, s_load path for indices
    const int wave = __builtin_amdgcn_readfirstlane((int)(threadIdx.x >> 5));
    const int m = blockIdx.x * WAVES_PER_BLOCK + wave;
    if (m >= M) return;  // uniform scalar branch; EXEC all-ones for WMMA

    const long long start = splits[m];
    const long long end   = splits[m + 1];
    const long long cnt   = end - start;
    const float w = 1.0f / (float)(cnt > 0 ? cnt : 1);

    const int half = lane >> 4;   // 0 -> chunk neighbors {0,1}; 1 -> {2,3}
    const int f    = lane & 15;   // feature within 16-block == A-matrix row M

    v8f acc0 = {}, acc1 = {}, acc2 = {}, acc3 = {};
    const v2f bmat = { w, w };    // B[4,16] all equal -> layout-independent

    // Load one chunk of 4 neighbor rows into A-fragments. 32-bit element
    // offsets (table is 2^26 elements) -> SADDR-form b32 gathers, imm offsets
    // 0/64/128/192. Each load: 16 lanes x 4B = 64B contiguous; 8 loads = 4
    // full 256B neighbor rows (exactly the required traffic).
    auto load_chunk = [&](long long e, v2f& c0, v2f& c1, v2f& c2, v2f& c3) {
        const long long i0 = nbr_idx[e];        // uniform s_load (mergeable)
        const long long i1 = nbr_idx[e + 1];
        const long long i2 = nbr_idx[e + 2];
        const long long i3 = nbr_idx[e + 3];
        const unsigned u0 = (unsigned)i0 << 6;  // SALU row offsets (elements)
        const unsigned u1 = (unsigned)i1 << 6;
        const unsigned u2 = (unsigned)i2 << 6;
        const unsigned u3 = (unsigned)i3 << 6;
        const unsigned oa = (half ? u2 : u0) + f;  // 1 cndmask + 1 add
        const unsigned ob = (half ? u3 : u1) + f;
        c0 = (v2f){ feats[oa],      feats[ob]      };
        c1 = (v2f){ feats[oa + 16], feats[ob + 16] };
        c2 = (v2f){ feats[oa + 32], feats[ob + 32] };
        c3 = (v2f){ feats[oa + 48], feats[ob + 48] };
    };

    // ---- main loop over full chunks, software double-buffered ----
    const long long nc = cnt >> 2;  // number of full chunks of 4
    if (nc > 0) {
        v2f a0, a1, a2, a3;
        load_chunk(start, a0, a1, a2, a3);
        for (long long c = 1; c < nc; ++c) {
            v2f b0, b1, b2, b3;
            load_chunk(start + 4 * c, b0, b1, b2, b3);  // issue next chunk
            WMMA4(a0, a1, a2, a3);                      // consume prev chunk
            a0 = b0; a1 = b1; a2 = b2; a3 = b3;
        }
        WMMA4(a0, a1, a2, a3);
    }

    // ---- tail: 1..3 remaining neighbors (never taken for uniform K=32) ----
    long long e = start + 4 * nc;
    if (e < end) {
        const long long i0 = nbr_idx[e];
        const long long i1 = (e + 1 < end) ? nbr_idx[e + 1] : -1ll;
        const long long i2 = (e + 2 < end) ? nbr_idx[e + 2] : -1ll;
        // e+3 < end would contradict chunking, so slot 3 is always empty.

        const long long na = half ? i2 : i0;
        const long long nb = half ? -1ll : i1;
        const float ma = (na >= 0) ? 1.0f : 0.0f;
        const float mb = (nb >= 0) ? 1.0f : 0.0f;
        const unsigned oa = ((unsigned)(na >= 0 ? na : 0) << 6) + f;
        const unsigned ob = ((unsigned)(nb >= 0 ? nb : 0) << 6) + f;

        v2f t0 = { feats[oa]      * ma, feats[ob]      * mb };
        v2f t1 = { feats[oa + 16] * ma, feats[ob + 16] * mb };
        v2f t2 = { feats[oa + 32] * ma, feats[ob + 32] * mb };
        v2f t3 = { feats[oa + 48] * ma, feats[ob + 48] * mb };
        WMMA4(t0, t1, t2, t3);
    }

    // D layout: VGPR r holds M=r (lanes 0-15) and M=r+8 (lanes 16-31); all
    // 16 columns identical (B columns equal). Feature b*16+g lives in element
    // g of lane 0 (g<8) or element g-8 of lane 16 (g>=8). Non-temporal: the
    // 32 MB output is write-once; keep it out of L2 so gather reuse survives.
    if ((lane & 15) == 0) {
        float* po = out + (long long)m * 64 + half * 8;
        const v8f ac[4] = { acc0, acc1, acc2, acc3 };
#pragma unroll
        for (int b = 0; b < 4; ++b) {
            v4f lo = { ac[b][0], ac[b][1], ac[b][2], ac[b][3] };
            v4f hi = { ac[b][4], ac[b][5], ac[b][6], ac[b][7] };
            __builtin_nontemporal_store(lo, (v4f*)(po + b * 16));      // b128
            __builtin_nontemporal_store(hi, (v4f*)(po + b * 16 + 4));  // b128
        }
    }
}

extern "C" void kernel_launch(void* const* d_in, const int* in_sizes, int n_in,
                              void* d_out, int out_size, void* d_ws, size_t ws_size,
                              hipStream_t stream) {
    const float*     feats  = (const float*)d_in[0];      // [N*64] f32
    const long long* nidx   = (const long long*)d_in[1];  // [E] i64
    const long long* splits = (const long long*)d_in[2];  // [M+1] i64
    float*           out    = (float*)d_out;              // [M*64] f32

    const int M = in_sizes[2] - 1;
    const int blocks = (M + WAVES_PER_BLOCK - 1) / WAVES_PER_BLOCK;
    neighbor_pool_wmma<<<blocks, 256, 0, stream>>>(feats, nidx, splits, out, M);
}